// ADMMSolver_85899346523
// MI455X (gfx1250) — compile-verified
//
#include <hip/hip_runtime.h>

typedef __attribute__((ext_vector_type(2))) float v2f;
typedef __attribute__((ext_vector_type(8))) float v8f;

#define XDIM   256
#define NIQ    128
#define NDIM   384   // XDIM + NIQ
#define BATCH  256
#define NSTEPS 5

// ---- workspace layout (float offsets) ----
#define WS_QH   0                          // 256*256  (Q + I, then scratch)
#define WS_PINV (WS_QH   + XDIM * XDIM)    // 256*256
#define WS_AP   (WS_PINV + XDIM * XDIM)    // 128*256  A @ Pinv
#define WS_S    (WS_AP   + NIQ * XDIM)     // 128*128  AP@A^T + I
#define WS_SINV (WS_S    + NIQ * NIQ)      // 128*128
#define WS_UT   (WS_SINV + NIQ * NIQ)      // 128*256  Sinv @ AP
#define WS_T    (WS_UT   + NIQ * XDIM)     // 384*512  T = Kinv[:384,:]
#define WS_TC   (WS_T    + NDIM * 512)     // 384*384
#define WS_BS   (WS_TC   + NDIM * NDIM)    // 384*256
#define WS_MZ   (WS_BS   + NDIM * XDIM)    // 384*384
#define WS_MP   (WS_MZ   + NDIM * NDIM)    // 384*384
#define WS_Y0   (WS_MP   + NDIM * NDIM)    // 256*384 x4 ping-pong
#define WS_W0   (WS_Y0   + BATCH * NDIM)
#define WS_Y1   (WS_W0   + BATCH * NDIM)
#define WS_W1   (WS_Y1   + BATCH * NDIM)

// ---- d_out layout (floats): x_out | rgap | sgap | x_hist ----
#define OUT_XOUT 0
#define OUT_RGAP (BATCH * XDIM)
#define OUT_SGAP (OUT_RGAP + BATCH * NDIM)
#define OUT_HIST (OUT_SGAP + BATCH * NDIM)

__device__ __forceinline__ v8f wmma4(v2f a, v2f b, v8f c) {
  // V_WMMA_F32_16X16X4_F32 : D = A(16x4) * B(4x16) + C(16x16), fp32 tensor path
  return __builtin_amdgcn_wmma_f32_16x16x4_f32(
      false, a, false, b, (short)0, c, false, false);
}

// ---------------- 1) Qh = Q + I ----------------
__global__ void build_qh_kernel(const float* __restrict__ Q, float* __restrict__ Qh) {
  int i = blockIdx.x * 256 + threadIdx.x;  // < 256*256
  int r = i >> 8, c = i & 255;
  Qh[i] = Q[i] + ((r == c) ? 1.0f : 0.0f);
}

// ------- 2) in-place Gauss-Jordan inversion, matrix resident in LDS --------
// sweep form: p=1/a_kk; row_k*=p (a_kk=p); a_ij-=a_ik*row_k (i!=k,j!=k);
//             col_k(i!=k) = -a_ik*p.  SPD inputs => no pivoting needed.
__global__ __launch_bounds__(1024) void gj_lds_kernel(const float* __restrict__ In,
                                                      float* __restrict__ Out,
                                                      int n, int shift) {
  extern __shared__ float Msh[];        // n*n floats (<=256KB of 320KB LDS)
  __shared__ float prow[256];
  __shared__ float pcol[256];
  __shared__ float pinv_s;
  int tid = threadIdx.x;
  int total = n * n;
  for (int i = tid; i < total; i += 1024) Msh[i] = In[i];
  __syncthreads();
  for (int k = 0; k < n; ++k) {
    if (tid == 0) pinv_s = 1.0f / Msh[k * n + k];
    __syncthreads();
    float p = pinv_s;
    if (tid < n) {
      int j = tid;
      float v = Msh[k * n + j] * p;          // scale pivot row
      prow[j] = (j == k) ? 0.0f : v;
      pcol[j] = (j == k) ? 0.0f : Msh[j * n + k];
      Msh[k * n + j] = (j == k) ? p : v;
    }
    __syncthreads();
    for (int idx = tid; idx < total; idx += 1024) {
      int i = idx >> shift;
      int j = idx & (n - 1);
      float v = Msh[idx] - pcol[i] * prow[j];
      if (j == k && i != k) v = -pcol[i] * p;
      if (i != k) Msh[idx] = v;              // row k already final
    }
    __syncthreads();
  }
  for (int i = tid; i < total; i += 1024) Out[i] = Msh[i];
}

// ---------------- 3) AP = A(128x256) @ Pinv(256x256)  [Pinv symmetric] -----
__global__ __launch_bounds__(256) void gemm_ap_kernel(const float* __restrict__ A,
                                                      const float* __restrict__ Pinv,
                                                      float* __restrict__ AP) {
  int wid = threadIdx.x >> 5, lane = threadIdx.x & 31;
  int lo = lane & 15, hi = lane >> 4;
  int tile = blockIdx.x * 8 + wid;        // 0..127 = 8(m) x 16(n)
  int tn = tile & 15, tm = tile >> 4;
  int mbase = tm * 16, nbase = tn * 16;
  v8f c = {};
  int arow = (mbase + lo) * XDIM;
  int brow = (nbase + lo) * XDIM;         // B[k][n] = Pinv[k][n] = Pinv[n][k]
  for (int k = 0; k < XDIM; k += 4) {
    int k0 = k + hi * 2;
    v2f a = *(const v2f*)(A + arow + k0);
    v2f b = *(const v2f*)(Pinv + brow + k0);
    c = wmma4(a, b, c);
  }
#pragma unroll
  for (int v = 0; v < 8; ++v)
    AP[(mbase + v + 8 * hi) * XDIM + nbase + lo] = c[v];
}

// ---------------- 4) S = AP(128x256) @ A^T + I -----------------------------
__global__ __launch_bounds__(256) void gemm_s_kernel(const float* __restrict__ AP,
                                                     const float* __restrict__ A,
                                                     float* __restrict__ S) {
  int wid = threadIdx.x >> 5, lane = threadIdx.x & 31;
  int lo = lane & 15, hi = lane >> 4;
  int tile = blockIdx.x * 8 + wid;        // 0..63 = 8(m) x 8(n)
  int tn = tile & 7, tm = tile >> 3;
  int mbase = tm * 16, nbase = tn * 16;
  v8f c = {};
  int arow = (mbase + lo) * XDIM;
  int brow = (nbase + lo) * XDIM;         // B[k][n] = A^T[k][n] = A[n][k]
  for (int k = 0; k < XDIM; k += 4) {
    int k0 = k + hi * 2;
    v2f a = *(const v2f*)(AP + arow + k0);
    v2f b = *(const v2f*)(A + brow + k0);
    c = wmma4(a, b, c);
  }
#pragma unroll
  for (int v = 0; v < 8; ++v) {
    int m = mbase + v + 8 * hi, n = nbase + lo;
    S[m * NIQ + n] = c[v] + ((m == n) ? 1.0f : 0.0f);
  }
}

// ---------------- 5) Ut = Sinv(128x128) @ AP(128x256) ----------------------
__global__ __launch_bounds__(256) void gemm_ut_kernel(const float* __restrict__ Sinv,
                                                      const float* __restrict__ AP,
                                                      float* __restrict__ Ut) {
  int wid = threadIdx.x >> 5, lane = threadIdx.x & 31;
  int lo = lane & 15, hi = lane >> 4;
  int tile = blockIdx.x * 8 + wid;        // 0..127 = 8(m) x 16(n)
  int tn = tile & 15, tm = tile >> 4;
  int mbase = tm * 16, nbase = tn * 16;
  v8f c = {};
  int arow = (mbase + lo) * NIQ;
  for (int k = 0; k < NIQ; k += 4) {
    int k0 = k + hi * 2;
    v2f a = *(const v2f*)(Sinv + arow + k0);
    v2f b;
    b.x = AP[k0 * XDIM + nbase + lo];     // B[k][n] = AP[k][n]
    b.y = AP[(k0 + 1) * XDIM + nbase + lo];
    c = wmma4(a, b, c);
  }
#pragma unroll
  for (int v = 0; v < 8; ++v)
    Ut[(mbase + v + 8 * hi) * XDIM + nbase + lo] = c[v];
}

// -------- 6) T[:, :256] = [Pinv;0] - U @ AP   (U = [Ut^T ; Sinv]) ----------
__global__ __launch_bounds__(256) void gemm_tleft_kernel(const float* __restrict__ Ut,
                                                         const float* __restrict__ Sinv,
                                                         const float* __restrict__ AP,
                                                         const float* __restrict__ Pinv,
                                                         float* __restrict__ T) {
  int wid = threadIdx.x >> 5, lane = threadIdx.x & 31;
  int lo = lane & 15, hi = lane >> 4;
  int tile = blockIdx.x * 8 + wid;        // 0..383 = 24(m) x 16(n)
  int tn = tile & 15, tm = tile >> 4;
  int mbase = tm * 16, nbase = tn * 16;
  v8f c = {};
  for (int k = 0; k < NIQ; k += 4) {
    int k0 = k + hi * 2;
    v2f a;
    if (mbase < XDIM) {                   // U[m][k] = Ut[k][m]
      a.x = Ut[k0 * XDIM + mbase + lo];
      a.y = Ut[(k0 + 1) * XDIM + mbase + lo];
    } else {                              // U[m][k] = Sinv[m-256][k]
      a = *(const v2f*)(Sinv + (mbase + lo - XDIM) * NIQ + k0);
    }
    v2f b;
    b.x = AP[k0 * XDIM + nbase + lo];     // B[k][n] = AP[k][n]
    b.y = AP[(k0 + 1) * XDIM + nbase + lo];
    c = wmma4(a, b, c);
  }
#pragma unroll
  for (int v = 0; v < 8; ++v) {
    int m = mbase + v + 8 * hi, n = nbase + lo;
    float base = (m < XDIM) ? Pinv[m * XDIM + n] : 0.0f;
    T[m * 512 + n] = base - c[v];
  }
}

// -------- 7) T[:, 256:384] = [0;I] - U ;  T[:, 384:512] = U ---------------
__global__ void pack_tright_kernel(const float* __restrict__ Ut,
                                   const float* __restrict__ Sinv,
                                   float* __restrict__ T) {
  int i = blockIdx.x * 256 + threadIdx.x;  // < 384*256
  int r = i >> 8, c2 = i & 255;
  int m = (c2 < NIQ) ? c2 : (c2 - NIQ);
  float u = (r < XDIM) ? Ut[m * XDIM + r] : Sinv[(r - XDIM) * NIQ + m];
  float v;
  if (c2 < NIQ) v = (((r - XDIM) == m) ? 1.0f : 0.0f) - u;  // cols 256..383
  else          v = u;                                       // cols 384..511
  T[r * 512 + XDIM + c2] = v;
}

// -------- 8) pack Tc, Bs, Mp, and slack columns of Mz ----------------------
__global__ void pack_kernel(const float* __restrict__ T,
                            const float* __restrict__ Q,
                            const float* __restrict__ A,
                            float* __restrict__ Tc, float* __restrict__ Bs,
                            float* __restrict__ Mp, float* __restrict__ Mz) {
  int c = blockIdx.x * 128 + threadIdx.x; // 0..383
  int r = blockIdx.y;                     // 0..383
  if (c >= NDIM) return;
  const float* Trow = T + r * 512;
  int src = (c < XDIM) ? c : c + NIQ;     // skip dual-block columns
  float t = Trow[src];
  Tc[r * NDIM + c] = t;                   // [T[:, :256], T[:,384:512]]
  Mp[r * NDIM + c] = (c < XDIM) ? -t : t; // Mp closed form
  if (c >= XDIM)                          // Mz slack cols: I - T[:,384:]
    Mz[r * NDIM + c] = ((r == c) ? 1.0f : 0.0f) - t;
  if (c < XDIM)                           // stacked [Q; A]
    Bs[r * XDIM + c] = (r < XDIM) ? Q[r * XDIM + c] : A[(r - XDIM) * XDIM + c];
}

// -------- 9) Mz[:, :256] = I - Tc @ Bs  (WMMA f32) -------------------------
__global__ __launch_bounds__(256) void mz_gemm_kernel(const float* __restrict__ Tc,
                                                      const float* __restrict__ Bs,
                                                      float* __restrict__ Mz) {
  int wid = threadIdx.x >> 5, lane = threadIdx.x & 31;
  int lo = lane & 15, hi = lane >> 4;
  int tile = blockIdx.x * 8 + wid;   // 0..383 = 24(r) x 16(j)
  int tj = tile & 15, tr = tile >> 4;
  int rbase = tr * 16, jbase = tj * 16;
  v8f c = {};
  int arow = (rbase + lo) * NDIM;
  for (int k = 0; k < NDIM; k += 4) {
    int k0 = k + hi * 2;
    v2f a = *(const v2f*)(Tc + arow + k0);
    v2f b;
    b.x = Bs[k0 * XDIM + jbase + lo];
    b.y = Bs[(k0 + 1) * XDIM + jbase + lo];
    c = wmma4(a, b, c);
  }
#pragma unroll
  for (int v = 0; v < 8; ++v) {
    int r = rbase + v + 8 * hi, j = jbase + lo;
    Mz[r * NDIM + j] = ((r == j) ? 1.0f : 0.0f) - c[v];
  }
}

// -------- 10) init: Y0 = [x, 0], W0 = 0, hist slot 0 -----------------------
__global__ void init_kernel(const float* __restrict__ x,
                            float* __restrict__ Y0, float* __restrict__ W0,
                            float* __restrict__ out) {
  int i = blockIdx.x * 256 + threadIdx.x;  // < BATCH*NDIM
  int b = i / NDIM, n = i - b * NDIM;
  float v = (n < XDIM) ? x[b * XDIM + n] : 0.0f;
  Y0[i] = v;
  W0[i] = 0.0f;
  out[OUT_HIST + i] = v;
}

// -------- 11) fused ADMM step: XK = (Y-W)Mz^T + P Mp^T + epilogue ----------
__global__ __launch_bounds__(256) void admm_step_kernel(
    const float* __restrict__ Yin, const float* __restrict__ Win,
    float* __restrict__ Yout, float* __restrict__ Wout,
    const float* __restrict__ parms, const float* __restrict__ Mz,
    const float* __restrict__ Mp, float* __restrict__ out, int step) {
  int wid = threadIdx.x >> 5, lane = threadIdx.x & 31;
  int lo = lane & 15, hi = lane >> 4;
  int tile = blockIdx.x * 8 + wid;   // 0..383 = 16(batch) x 24(nd)
  int tr = tile & 15, tc = tile >> 4;
  int rbase = tr * 16, nbase = tc * 16;

  v8f c = {};
  int arow = (rbase + lo) * NDIM;    // A fragments: batch rows
  int brow = (nbase + lo) * NDIM;    // B fragments: Mz/Mp rows, K-contiguous
#pragma unroll 2
  for (int k = 0; k < NDIM; k += 4) {
    int k0 = k + hi * 2;
    v2f y2 = *(const v2f*)(Yin + arow + k0);
    v2f w2 = *(const v2f*)(Win + arow + k0);
    v2f az = y2 - w2;                                    // Z = Y - W on the fly
    v2f bz = *(const v2f*)(Mz + brow + k0);
    c = wmma4(az, bz, c);
    v2f ap = *(const v2f*)(parms + arow + k0);
    v2f bp = *(const v2f*)(Mp + brow + k0);
    c = wmma4(ap, bp, c);
  }

  float* hist = out + OUT_HIST + (size_t)(step + 1) * BATCH * NDIM;
#pragma unroll
  for (int v = 0; v < 8; ++v) {
    int b = rbase + v + 8 * hi;
    int n = nbase + lo;
    int idx = b * NDIM + n;
    float xk = c[v];                 // xA == xk since 2*ALPHA == 1
    float yold = Yin[idx], wold = Win[idx];
    float lowern = (n < XDIM) ? -1000.0f : 0.0f;
    float yn = fminf(fmaxf(xk + wold, lowern), 1000.0f);
    float wn = wold + xk - yn;
    Yout[idx] = yn;
    Wout[idx] = wn;
    hist[idx] = xk;
    if (step == NSTEPS - 1) {
      out[OUT_RGAP + idx] = xk - yn;
      out[OUT_SGAP + idx] = yn - yold;
      if (n < XDIM) out[OUT_XOUT + b * XDIM + n] = xk;
    }
  }
}

extern "C" void kernel_launch(void* const* d_in, const int* in_sizes, int n_in,
                              void* d_out, int out_size, void* d_ws, size_t ws_size,
                              hipStream_t stream) {
  const float* x     = (const float*)d_in[0];
  const float* parms = (const float*)d_in[1];
  const float* Q     = (const float*)d_in[2];
  const float* A     = (const float*)d_in[3];
  float* out = (float*)d_out;
  float* ws  = (float*)d_ws;

  float* Qh   = ws + WS_QH;
  float* Pinv = ws + WS_PINV;
  float* AP   = ws + WS_AP;
  float* S    = ws + WS_S;
  float* Sinv = ws + WS_SINV;
  float* Ut   = ws + WS_UT;
  float* T    = ws + WS_T;
  float* Tc   = ws + WS_TC;
  float* Bs   = ws + WS_BS;
  float* Mz   = ws + WS_MZ;
  float* Mp   = ws + WS_MP;
  float* Yb[2] = {ws + WS_Y0, ws + WS_Y1};
  float* Wb[2] = {ws + WS_W0, ws + WS_W1};

  // --- operator construction: LDS-resident inversions + WMMA GEMMs ---
  build_qh_kernel<<<256, 256, 0, stream>>>(Q, Qh);
  gj_lds_kernel<<<1, 1024, XDIM * XDIM * sizeof(float), stream>>>(Qh, Pinv, XDIM, 8);
  gemm_ap_kernel<<<16, 256, 0, stream>>>(A, Pinv, AP);
  gemm_s_kernel<<<8, 256, 0, stream>>>(AP, A, S);
  gj_lds_kernel<<<1, 1024, NIQ * NIQ * sizeof(float), stream>>>(S, Sinv, NIQ, 7);
  gemm_ut_kernel<<<16, 256, 0, stream>>>(Sinv, AP, Ut);
  gemm_tleft_kernel<<<48, 256, 0, stream>>>(Ut, Sinv, AP, Pinv, T);
  pack_tright_kernel<<<384, 256, 0, stream>>>(Ut, Sinv, T);
  pack_kernel<<<dim3(3, NDIM), 128, 0, stream>>>(T, Q, A, Tc, Bs, Mp, Mz);
  mz_gemm_kernel<<<48, 256, 0, stream>>>(Tc, Bs, Mz);

  // --- ADMM iterations ---
  init_kernel<<<(BATCH * NDIM) / 256, 256, 0, stream>>>(x, Yb[0], Wb[0], out);
  for (int s = 0; s < NSTEPS; ++s) {
    admm_step_kernel<<<48, 256, 0, stream>>>(Yb[s & 1], Wb[s & 1],
                                             Yb[(s + 1) & 1], Wb[(s + 1) & 1],
                                             parms, Mz, Mp, out, s);
  }
}